// GNNModel_42064909697775
// MI455X (gfx1250) — compile-verified
//
#include <hip/hip_runtime.h>
#include <hip/hip_bf16.h>

typedef __attribute__((ext_vector_type(2))) float v2f;
typedef __attribute__((ext_vector_type(8))) float v8f;

#define HDIM 128

// ---------------------------------------------------------------- utilities
__global__ void zero_f32(float* __restrict__ p, long n) {
  long i = (long)blockIdx.x * blockDim.x + threadIdx.x;
  long stride = (long)gridDim.x * blockDim.x;
  for (; i < n; i += stride) p[i] = 0.0f;
}

// deg[dst] += w  (hardware global_atomic_add_f32)
__global__ void deg_kernel(const int* __restrict__ dst,
                           const float* __restrict__ w,
                           float* __restrict__ deg, int E) {
  int i = blockIdx.x * blockDim.x + threadIdx.x;
  int stride = gridDim.x * blockDim.x;
  for (; i < E; i += stride) unsafeAtomicAdd(&deg[dst[i]], w[i]);
}

// in place: deg -> rsqrt(deg + 1)   (+1 = self loop)
__global__ void dinv_kernel(float* __restrict__ deg, int n) {
  int i = blockIdx.x * blockDim.x + threadIdx.x;
  int stride = gridDim.x * blockDim.x;
  for (; i < n; i += stride) deg[i] = rsqrtf(deg[i] + 1.0f);
}

// ------------------------------------------------------------- f32 WMMA GEMM
// C[N x 128] = A[N x 128] @ W[128 x 128], exact f32 via V_WMMA_F32_16X16X4_F32.
// Block = 256 threads = 8 waves; wave w owns column tile w (16 cols).
// Each wave preloads its 32 B-fragments (all of K) once, then grid-strides
// over 16-row tiles: 32 wmma per tile.
__global__ void __launch_bounds__(256, 2)
gemm_wmma_f32(const float* __restrict__ A, const float* __restrict__ W,
              float* __restrict__ C, int nRowTiles) {
  const int lane = threadIdx.x & 31;
  const int wave = threadIdx.x >> 5;   // 0..7 -> column tile
  const int m    = lane & 15;          // row (A) / col (B,C) within tile
  const int half = lane >> 4;          // K-half selector
  const int col0 = wave * 16;

  // B fragment layout (4x16, f32): VGPR j holds K = 2*half + j, col = m.
  v2f bfrag[32];
#pragma unroll
  for (int kk = 0; kk < 32; ++kk) {
    const int k = kk * 4 + half * 2;
    v2f b;
    b.x = W[(k + 0) * HDIM + col0 + m];
    b.y = W[(k + 1) * HDIM + col0 + m];
    bfrag[kk] = b;
  }

  for (int rt = blockIdx.x; rt < nRowTiles; rt += gridDim.x) {
    const float* __restrict__ arow = A + (long)(rt * 16 + m) * HDIM;
    v8f acc = {0.f, 0.f, 0.f, 0.f, 0.f, 0.f, 0.f, 0.f};
#pragma unroll
    for (int kk = 0; kk < 32; ++kk) {
      const int k = kk * 4 + half * 2;
      // A fragment (16x4, f32): lane m holds row m, VGPR j holds K=2*half+j
      const v2f a = *(const v2f*)(arow + k);   // 8B-aligned contiguous pair
      acc = __builtin_amdgcn_wmma_f32_16x16x4_f32(
          /*neg_a=*/false, a, /*neg_b=*/false, bfrag[kk],
          /*c_mod=*/(short)0, acc, /*reuse_a=*/false, /*reuse_b=*/false);
    }
    // C layout: VGPR r holds row = 8*half + r, col = col0 + m
#pragma unroll
    for (int r = 0; r < 8; ++r)
      C[(long)(rt * 16 + half * 8 + r) * HDIM + col0 + m] = acc[r];
  }
}

// ------------------------------------------------------- edge scatter (L2)
// One wave per edge: lane loads float4 of h[src] (512B per wave, L2-resident),
// scales by norm = dinv[src]*w*dinv[dst], 4 hardware f32 atomics into agg[dst].
__global__ void __launch_bounds__(256)
scatter_kernel(const float* __restrict__ h, const float* __restrict__ dinv,
               const int* __restrict__ src, const int* __restrict__ dst,
               const float* __restrict__ w, float* __restrict__ agg, int E) {
  const int lane = threadIdx.x & 31;
  int wid = blockIdx.x * (blockDim.x >> 5) + (threadIdx.x >> 5);
  const int nwaves = gridDim.x * (blockDim.x >> 5);
  for (int e = wid; e < E; e += nwaves) {
    const int s = src[e];
    const int d = dst[e];
    const float norm = dinv[s] * w[e] * dinv[d];
    const float4 hv = *(const float4*)(h + (long)s * HDIM + lane * 4);
    float* __restrict__ out = agg + (long)d * HDIM + lane * 4;
    unsafeAtomicAdd(out + 0, norm * hv.x);
    unsafeAtomicAdd(out + 1, norm * hv.y);
    unsafeAtomicAdd(out + 2, norm * hv.z);
    unsafeAtomicAdd(out + 3, norm * hv.w);
  }
}

// io = (relu?) (io + dinv^2 * h + b)     (self-loop term + bias, in place)
__global__ void finalize_kernel(float* __restrict__ io,
                                const float* __restrict__ h,
                                const float* __restrict__ dinv,
                                const float* __restrict__ b,
                                long total, int relu) {
  long i = (long)blockIdx.x * blockDim.x + threadIdx.x;
  long stride = (long)gridDim.x * blockDim.x;
  for (; i < total; i += stride) {
    const int node = (int)(i >> 7);
    const int f = (int)(i & (HDIM - 1));
    const float di = dinv[node];
    float v = io[i] + di * di * h[i] + b[f];
    io[i] = relu ? fmaxf(v, 0.0f) : v;
  }
}

// ----------------------------------------------------------------- launcher
extern "C" void kernel_launch(void* const* d_in, const int* in_sizes, int n_in,
                              void* d_out, int out_size, void* d_ws, size_t ws_size,
                              hipStream_t stream) {
  const float* x  = (const float*)d_in[0];
  const float* ew = (const float*)d_in[1];
  const float* W1 = (const float*)d_in[2];
  const float* b1 = (const float*)d_in[3];
  const float* W2 = (const float*)d_in[4];
  const float* b2 = (const float*)d_in[5];
  const int*   ei = (const int*)d_in[6];

  const int N = in_sizes[0] / HDIM;   // 100000
  const int E = in_sizes[1];          // 1600000
  const int* src = ei;                // edge_index[0]
  const int* dst = ei + E;            // edge_index[1]

  const long NH = (long)N * HDIM;

  // Workspace layout: [deg/dinv: N][h: N*H][agg: N*H]  (~103 MB)
  float* deg = (float*)d_ws;
  float* h   = deg + (((long)N + 255) & ~255L);
  float* agg = h + NH;
  float* out = (float*)d_out;

  // init accumulators (ws/d_out are poisoned by harness)
  zero_f32<<<512, 256, 0, stream>>>(deg, N);
  zero_f32<<<2048, 256, 0, stream>>>(agg, NH);
  zero_f32<<<2048, 256, 0, stream>>>(out, NH);

  // symmetric normalization coefficients
  deg_kernel<<<2048, 256, 0, stream>>>(dst, ew, deg, E);
  dinv_kernel<<<512, 256, 0, stream>>>(deg, N);   // deg -> dinv in place

  const int nRowTiles = N / 16;       // N is a multiple of 16

  // ---- layer 1: h = x @ W1 ; agg = scatter ; h1r = relu(agg + dinv^2*h + b1)
  gemm_wmma_f32<<<1024, 256, 0, stream>>>(x, W1, h, nRowTiles);
  scatter_kernel<<<4096, 256, 0, stream>>>(h, deg, src, dst, ew, agg, E);
  finalize_kernel<<<2048, 256, 0, stream>>>(agg, h, deg, b1, NH, 1);

  // ---- layer 2: h2 = h1r @ W2 ; scatter into d_out ; add self-loop + b2
  gemm_wmma_f32<<<1024, 256, 0, stream>>>(agg, W2, h, nRowTiles);
  scatter_kernel<<<4096, 256, 0, stream>>>(h, deg, src, dst, ew, out, E);
  finalize_kernel<<<2048, 256, 0, stream>>>(out, h, deg, b2, NH, 0);
}